// RadonTransform_56504589746205
// MI455X (gfx1250) — compile-verified
//
#include <hip/hip_runtime.h>

// Radon transform, fused rotate + row-sum (sinogram).
// H=W=512, DIAG=725, PAD=106, P=724, 180 angles (n*pi/179 rad).
// d_out = [ sinogram 180*724 | rotated 180*724*724 ] float32.
//
// Fast path: prologue builds a 1028x1028 zero-padded source in d_ws so
// bilinear taps need no clamps/masks: one 32-bit byte offset + paired b64
// loads (immediate offset for the second row). One wave32 per (angle,
// 16-row strip); 181 slabs of 4 columns; each lane computes 2 samples = its
// two A VGPRs of V_WMMA_F32_16X16X4_F32; WMMA with B=ones accumulates the 16
// per-row sums. Two accumulators (even/odd slabs) avoid D->C register copies.
// Rotated pixels stream out with non-temporal b64 stores.

typedef __attribute__((ext_vector_type(2))) float v2f;
typedef __attribute__((ext_vector_type(8))) float v8f;

#define RT_P      724
#define RT_SRC    512
#define RT_NANG   180
#define RT_STRIPS 46                   // ceil(724/16)
#define RT_SLABS  181                  // 724 / 4
#define RT_CENTER 361.5f               // (P-1)/2

// Big padded buffer: sample coords ix,iy in [-150.5, 874.5]; +152 shift ->
// tap indices in [1, 1026] subset of [0, 1028). Image pixel (r,c) lives at
// big[r+258][c+258].
#define RT_BW     1028
#define RT_BOFF   152
#define RT_IOFF   258
#define RT_WS_NEED ((size_t)RT_BW * RT_BW * 4 + 4096)

// ---------------------------------------------------------------- prologue
__global__ __launch_bounds__(256) void radon_prep_kernel(
    const float* __restrict__ img, float* __restrict__ big)
{
    int idx = blockIdx.x * blockDim.x + threadIdx.x;
    if (idx >= RT_BW * RT_BW) return;
    int y = idx / RT_BW;
    int x = idx - y * RT_BW;
    int sy = y - RT_IOFF;
    int sx = x - RT_IOFF;
    float v = 0.0f;
    if ((unsigned)sy < (unsigned)RT_SRC && (unsigned)sx < (unsigned)RT_SRC)
        v = img[sy * RT_SRC + sx];
    big[idx] = v;
}

// ---------------------------------------------------------------- fast path
__device__ __forceinline__ float rt_bilinear_fast(const float* __restrict__ big,
                                                  float ixb, float iyb) {
    float fx = floorf(ixb);
    float fy = floorf(iyb);
    float wx = ixb - fx;
    float wy = iyb - fy;
    // Indices are guaranteed in [1, 1026]; use an unsigned byte offset so the
    // backend can keep a single 32-bit VGPR offset against the SGPR base.
    unsigned xi = (unsigned)(int)fx;
    unsigned yi = (unsigned)(int)fy;
    const float* p = (const float*)((const char*)big +
                                    (yi * (unsigned)(RT_BW * 4) + (xi << 2)));
    float v00 = p[0];
    float v01 = p[1];
    float v10 = p[RT_BW];
    float v11 = p[RT_BW + 1];
    float top = v00 + wx * (v01 - v00);
    float bot = v10 + wx * (v11 - v10);
    return top + wy * (bot - top);
}

__device__ __forceinline__ v2f rt_pair(const float* __restrict__ big,
                                       float u, float cs, float sn,
                                       float ix0, float iy0) {
    v2f a;
    a.x = rt_bilinear_fast(big, fmaf(cs, u, ix0),        fmaf(sn, u, iy0));
    a.y = rt_bilinear_fast(big, fmaf(cs, u + 1.0f, ix0), fmaf(sn, u + 1.0f, iy0));
    return a;
}

__global__ __launch_bounds__(256) void RadonTransform_fast_kernel(
    const float* __restrict__ big,   // [1028*1028] zero-padded source
    float* __restrict__ sino,        // [180*724]
    float* __restrict__ rot,         // [180*724*724]
    float* __restrict__ dump)        // scratch sink for rows >= 724
{
    const int wave = (int)((blockIdx.x * blockDim.x + threadIdx.x) >> 5);
    const int lane = (int)(threadIdx.x & 31u);
    if (wave >= RT_NANG * RT_STRIPS) return;   // uniform per wave

    const int n      = wave / RT_STRIPS;
    const int strip  = wave % RT_STRIPS;
    const int ybase  = strip * 16;
    const int row    = ybase + (lane & 15);
    const int colOff = (lane >> 4) * 2;        // K offset within 4-wide slab

    const float ang = (float)((double)n * (3.14159265358979323846 / 179.0));
    const float cs = cosf(ang);
    const float sn = sinf(ang);

    // Clamp the coordinate-generation row for the discarded rows of the last
    // strip so all sample coords stay inside the 1028x1028 buffer.
    const int rowC = min(row, RT_P - 1);
    const float v   = (float)rowC - RT_CENTER;
    const float ix0 = -sn * v + (RT_CENTER + (float)RT_BOFF);  // ixb = cs*u + ix0
    const float iy0 =  cs * v + (RT_CENTER + (float)RT_BOFF);  // iyb = sn*u + iy0

    const bool rowOK = (row < RT_P);
    float* __restrict__ sp =
        (rowOK ? (rot + ((size_t)(n * RT_P + row)) * RT_P) : dump) + colOff;

    v8f c0 = {};                                // even-slab row sums
    v8f c1 = {};                                // odd-slab row sums
    v2f bOnes = {1.0f, 1.0f};                   // B = ones (4x16)

    // u = column - 361.5; all values are k+0.5, stride 8.0 -> exact fp32.
    float u = (float)colOff - RT_CENTER;

    for (int i = 0; i < RT_SLABS / 2; ++i) {    // 90 slab pairs
        v2f a0 = rt_pair(big, u, cs, sn, ix0, iy0);
        __builtin_nontemporal_store(a0, (v2f*)sp);
        c0 = __builtin_amdgcn_wmma_f32_16x16x4_f32(
                 false, a0, false, bOnes, (short)0, c0, false, false);

        v2f a1 = rt_pair(big, u + 4.0f, cs, sn, ix0, iy0);
        __builtin_nontemporal_store(a1, (v2f*)(sp + 4));
        c1 = __builtin_amdgcn_wmma_f32_16x16x4_f32(
                 false, a1, false, bOnes, (short)0, c1, false, false);

        u += 8.0f;
        sp += 8;
    }
    {   // final slab (181st)
        v2f a0 = rt_pair(big, u, cs, sn, ix0, iy0);
        __builtin_nontemporal_store(a0, (v2f*)sp);
        c0 = __builtin_amdgcn_wmma_f32_16x16x4_f32(
                 false, a0, false, bOnes, (short)0, c0, false, false);
    }

    const v8f c = c0 + c1;

    // C VGPR k = rowsum(ybase+k) on lanes 0-15, rowsum(ybase+8+k) on 16-31.
    if (lane == 0 || lane == 16) {
        const int rbase = ybase + (lane >> 4) * 8;
#pragma unroll
        for (int k = 0; k < 8; ++k) {
            const int r = rbase + k;
            if (r < RT_P) sino[n * RT_P + r] = c[k];
        }
    }
}

// ------------------------------------------------- fallback (no ws needed)
#define RT_PAD 106
__device__ __forceinline__ float rt_tap(const float* __restrict__ img, int yi, int xi) {
    int sy = yi - RT_PAD;
    int sx = xi - RT_PAD;
    bool ok = ((unsigned)sy < (unsigned)RT_SRC) && ((unsigned)sx < (unsigned)RT_SRC);
    int cy = min(max(sy, 0), RT_SRC - 1);
    int cx = min(max(sx, 0), RT_SRC - 1);
    float val = img[cy * RT_SRC + cx];
    return ok ? val : 0.0f;
}

__device__ __forceinline__ float rt_bilinear_safe(const float* __restrict__ img,
                                                  float ix, float iy) {
    float fx = floorf(ix);
    float fy = floorf(iy);
    float wx = ix - fx;
    float wy = iy - fy;
    int x0 = (int)fx;
    int y0 = (int)fy;
    float v00 = rt_tap(img, y0,     x0);
    float v01 = rt_tap(img, y0,     x0 + 1);
    float v10 = rt_tap(img, y0 + 1, x0);
    float v11 = rt_tap(img, y0 + 1, x0 + 1);
    float top = v00 + wx * (v01 - v00);
    float bot = v10 + wx * (v11 - v10);
    return top + wy * (bot - top);
}

__global__ __launch_bounds__(256) void RadonTransform_safe_kernel(
    const float* __restrict__ img, float* __restrict__ sino, float* __restrict__ rot)
{
    const int wave = (int)((blockIdx.x * blockDim.x + threadIdx.x) >> 5);
    const int lane = (int)(threadIdx.x & 31u);
    if (wave >= RT_NANG * RT_STRIPS) return;

    const int n      = wave / RT_STRIPS;
    const int strip  = wave % RT_STRIPS;
    const int ybase  = strip * 16;
    const int row    = ybase + (lane & 15);
    const int colOff = (lane >> 4) * 2;

    const float ang = (float)((double)n * (3.14159265358979323846 / 179.0));
    const float cs = cosf(ang);
    const float sn = sinf(ang);

    const float v   = (float)row - RT_CENTER;
    const float ix0 = -sn * v + RT_CENTER;
    const float iy0 =  cs * v + RT_CENTER;

    const bool rowOK = (row < RT_P);
    float* __restrict__ rowPtr = rot + ((size_t)(n * RT_P + row)) * RT_P;

    v8f c = {};
    v2f bOnes = {1.0f, 1.0f};

    for (int slab = 0; slab < RT_SLABS; ++slab) {
        const int col0 = slab * 4 + colOff;
        const float u0 = (float)col0 - RT_CENTER;
        const float u1 = u0 + 1.0f;

        v2f a;
        a.x = rt_bilinear_safe(img, cs * u0 + ix0, sn * u0 + iy0);
        a.y = rt_bilinear_safe(img, cs * u1 + ix0, sn * u1 + iy0);

        if (rowOK) *(v2f*)(rowPtr + col0) = a;

        c = __builtin_amdgcn_wmma_f32_16x16x4_f32(
                false, a, false, bOnes, (short)0, c, false, false);
    }

    if (lane == 0 || lane == 16) {
        const int rbase = ybase + (lane >> 4) * 8;
#pragma unroll
        for (int k = 0; k < 8; ++k) {
            const int r = rbase + k;
            if (r < RT_P) sino[n * RT_P + r] = c[k];
        }
    }
}

// ---------------------------------------------------------------- launch
extern "C" void kernel_launch(void* const* d_in, const int* in_sizes, int n_in,
                              void* d_out, int out_size, void* d_ws, size_t ws_size,
                              hipStream_t stream) {
    (void)in_sizes; (void)n_in; (void)out_size;

    const float* img = (const float*)d_in[0];
    float* out  = (float*)d_out;
    float* sino = out;                              // [180*724]
    float* rot  = out + (size_t)RT_NANG * RT_P;     // [180*724*724]

    const int totalWaves = RT_NANG * RT_STRIPS;     // 8280 waves
    const int blockSize  = 256;                     // 8 waves per block
    const int grid       = (totalWaves * 32 + blockSize - 1) / blockSize;

    if (d_ws != nullptr && ws_size >= RT_WS_NEED) {
        float* big  = (float*)d_ws;                       // [1028*1028]
        float* dump = big + (size_t)RT_BW * RT_BW;        // sink for rows>=724
        const int prepN = RT_BW * RT_BW;
        radon_prep_kernel<<<(prepN + 255) / 256, 256, 0, stream>>>(img, big);
        RadonTransform_fast_kernel<<<grid, blockSize, 0, stream>>>(big, sino, rot, dump);
    } else {
        RadonTransform_safe_kernel<<<grid, blockSize, 0, stream>>>(img, sino, rot);
    }
}